// loss_6631429505127
// MI455X (gfx1250) — compile-verified
//
#include <hip/hip_runtime.h>
#include <hip/hip_bf16.h>
#include <math.h>

typedef __attribute__((ext_vector_type(2))) float v2f;
typedef __attribute__((ext_vector_type(8))) float v8f;

#define NPTS   16384
#define HDIM   4096
#define NMODES 1024
#define NNDIM  4096
#define REGUL_L2 1e-5f
#define CG_ITERS 512

// ---------------------------------------------------------------------------
// Native f32 WMMA: D(16x16) += A(16x4) * B(4x16), f32 end-to-end.
// A frag: lane L holds A[m = L%16][kbase + 2*(L>=16) + {0,1}] in {a.x, a.y}
// B frag: lane L holds B[kbase + 2*(L>=16) + {0,1}][n = L%16] in {b.x, b.y}
// C/D   : VGPR r, lanes 0-15 -> M=r, lanes 16-31 -> M=r+8 ; N = lane%16
// ---------------------------------------------------------------------------
#if defined(__HIP_DEVICE_COMPILE__) && __has_builtin(__builtin_amdgcn_wmma_f32_16x16x4_f32)
#define HAVE_WMMA_F32X4 1
#endif

__device__ __forceinline__ v8f wmma4(v2f a, v2f b, v8f c) {
#ifdef HAVE_WMMA_F32X4
  return __builtin_amdgcn_wmma_f32_16x16x4_f32(false, a, false, b, (short)0, c,
                                               false, false);
#else
  (void)a; (void)b;
  return c; // host pass / non-gfx1250 placeholder only
#endif
}

// ---------------------------------------------------------------------------
// Stage 1: h1 = tanh(pts*W1 + b1), dh1 = (1-h1^2)*W1   (elementwise)
// ---------------------------------------------------------------------------
__global__ void act_kernel(const float* __restrict__ pts,
                           const float* __restrict__ W1,
                           const float* __restrict__ b1,
                           float* __restrict__ h1, float* __restrict__ dh1) {
  long long idx = (long long)blockIdx.x * blockDim.x + threadIdx.x;
  int i = (int)(idx >> 12);          // HDIM == 4096 == 2^12
  int j = (int)(idx & (HDIM - 1));
  float w = W1[j];
  float h = tanhf(pts[i] * w + b1[j]);
  h1[idx]  = h;
  dh1[idx] = (1.f - h * h) * w;
}

// ---------------------------------------------------------------------------
// Stage 2 (fused): dh2 = (1 - tanh(h1@W2 + b2)^2) * (dh1@W2)
// Dual-stream WMMA GEMM: shares every B fragment between the two A streams.
// Block = 8 waves; wave computes a 16(M) x 64(N) strip for BOTH streams.
// ---------------------------------------------------------------------------
__global__ void __launch_bounds__(256)
gemm_dual_tanh(const float* __restrict__ H1, const float* __restrict__ G1,
               const float* __restrict__ W, const float* __restrict__ bias,
               float* __restrict__ Gout, int M, int N, int K) {
  const int lane = threadIdx.x & 31;
  const int wave = threadIdx.x >> 5;
  const int hi   = lane >> 4;        // 0..1 : lane half selects K pair
  const int l15  = lane & 15;
  const int bm   = blockIdx.y * 128 + wave * 16;
  const int bn   = blockIdx.x * 64;

  v8f accZ[4] = {};
  v8f accG[4] = {};
  const int mA = bm + l15;

  for (int k = 0; k < K; k += 4) {
    const int ka = k + (hi << 1);
    const float* pa1 = H1 + (size_t)mA * K + ka;
    const float* pa2 = G1 + (size_t)mA * K + ka;
    v2f a1; a1.x = pa1[0]; a1.y = pa1[1];
    v2f a2; a2.x = pa2[0]; a2.y = pa2[1];
#pragma unroll
    for (int t = 0; t < 4; ++t) {
      const float* pb = W + (size_t)ka * N + (bn + t * 16 + l15);
      v2f b; b.x = pb[0]; b.y = pb[N];
      accZ[t] = wmma4(a1, b, accZ[t]);
      accG[t] = wmma4(a2, b, accG[t]);
    }
  }
#pragma unroll
  for (int t = 0; t < 4; ++t) {
    const int n  = bn + t * 16 + l15;
    const float bb = bias[n];
#pragma unroll
    for (int r = 0; r < 8; ++r) {
      const int m = bm + (hi << 3) + r;
      float h = tanhf(accZ[t][r] + bb);
      Gout[(size_t)m * N + n] = (1.f - h * h) * accG[t][r];
    }
  }
}

// ---------------------------------------------------------------------------
// Generic WMMA GEMM: C = op(A) @ B [row-major], optional per-row scale and
// diagonal add (for AtA + lambda*I). TRANS_A: A_phys is (K x M), lda = M-dim.
// ---------------------------------------------------------------------------
template <bool TRANS_A, bool ROWSCALE, bool DIAG>
__global__ void __launch_bounds__(256)
gemm_wmma(const float* __restrict__ A, const float* __restrict__ B,
          float* __restrict__ C, const float* __restrict__ rowscale,
          float diag_add, int M, int N, int K, int lda, int ldb) {
  const int lane = threadIdx.x & 31;
  const int wave = threadIdx.x >> 5;
  const int hi   = lane >> 4;
  const int l15  = lane & 15;
  const int bm   = blockIdx.y * 128 + wave * 16;
  const int bn   = blockIdx.x * 64;

  v8f acc[4] = {};
  const int mA = bm + l15;

  for (int k = 0; k < K; k += 4) {
    const int ka = k + (hi << 1);
    v2f a;
    if (TRANS_A) {
      a.x = A[(size_t)ka * lda + mA];
      a.y = A[(size_t)(ka + 1) * lda + mA];
    } else {
      const float* pa = A + (size_t)mA * lda + ka;
      a.x = pa[0]; a.y = pa[1];
    }
#pragma unroll
    for (int t = 0; t < 4; ++t) {
      const float* pb = B + (size_t)ka * ldb + (bn + t * 16 + l15);
      v2f b; b.x = pb[0]; b.y = pb[ldb];
      acc[t] = wmma4(a, b, acc[t]);
    }
  }
#pragma unroll
  for (int t = 0; t < 4; ++t) {
    const int n = bn + t * 16 + l15;
#pragma unroll
    for (int r = 0; r < 8; ++r) {
      const int m = bm + (hi << 3) + r;
      float v = acc[t][r];
      if (ROWSCALE) v *= rowscale[m];
      if (DIAG && (m == n)) v += diag_add;
      C[(size_t)m * N + n] = v;
    }
  }
}

// ---------------------------------------------------------------------------
// rhs[n] = -sum_j matA[j][n] * vecB[j]
// ---------------------------------------------------------------------------
__global__ void rhs_kernel(const float* __restrict__ matA,
                           const float* __restrict__ vecB,
                           float* __restrict__ rhs) {
  const int n = blockIdx.x * blockDim.x + threadIdx.x;
  float s = 0.f;
  for (int j = 0; j < NMODES; ++j) s += matA[(size_t)j * NNDIM + n] * vecB[j];
  rhs[n] = -s;
}

// ---------------------------------------------------------------------------
// GEMV: y = A(MxN) @ x ; one row per wave, shuffle reduction (wave32).
// ---------------------------------------------------------------------------
__global__ void gemv_kernel(const float* __restrict__ A,
                            const float* __restrict__ x, float* __restrict__ y,
                            int M, int N) {
  const int lane = threadIdx.x & 31;
  const int row  = blockIdx.x * 8 + (threadIdx.x >> 5);
  if (row >= M) return;
  const float* ar = A + (size_t)row * N;
  float s = 0.f;
  for (int c = lane; c < N; c += 32) s += ar[c] * x[c];
#pragma unroll
  for (int o = 16; o > 0; o >>= 1) s += __shfl_down(s, o);
  if (lane == 0) y[row] = s;
}

// ---------------------------------------------------------------------------
// CG solve of (AtA) w = rhs ; deterministic single-block reductions.
// ---------------------------------------------------------------------------
__global__ void cg_init_kernel(const float* __restrict__ rhs, float* x,
                               float* r, float* p, float* sc, int n) {
  __shared__ float red[1024];
  const int tid = threadIdx.x;
  float rs = 0.f;
  for (int i = tid; i < n; i += 1024) {
    float v = rhs[i];
    x[i] = 0.f; r[i] = v; p[i] = v;
    rs += v * v;
  }
  red[tid] = rs; __syncthreads();
  for (int s = 512; s > 0; s >>= 1) {
    if (tid < s) red[tid] += red[tid + s];
    __syncthreads();
  }
  if (tid == 0) sc[0] = red[0];
}

__global__ void cg_step_kernel(const float* __restrict__ Ap, float* x, float* r,
                               float* p, float* sc, int n) {
  __shared__ float red[1024];
  const int tid = threadIdx.x;

  float pap = 0.f;
  for (int i = tid; i < n; i += 1024) pap += p[i] * Ap[i];
  red[tid] = pap; __syncthreads();
  for (int s = 512; s > 0; s >>= 1) {
    if (tid < s) red[tid] += red[tid + s];
    __syncthreads();
  }
  const float rsold = sc[0];
  const float alpha = rsold / red[0];
  __syncthreads();

  float rs = 0.f;
  for (int i = tid; i < n; i += 1024) {
    x[i] += alpha * p[i];
    float rn = r[i] - alpha * Ap[i];
    r[i] = rn;
    rs += rn * rn;
  }
  red[tid] = rs; __syncthreads();
  for (int s = 512; s > 0; s >>= 1) {
    if (tid < s) red[tid] += red[tid + s];
    __syncthreads();
  }
  const float rsnew = red[0];
  const float beta  = rsnew / rsold;
  for (int i = tid; i < n; i += 1024) p[i] = r[i] + beta * p[i];
  if (tid == 0) sc[0] = rsnew;
}

// ---------------------------------------------------------------------------
// loss = sum((FT + vecB)^2)
// ---------------------------------------------------------------------------
__global__ void loss_kernel(const float* __restrict__ FT,
                            const float* __restrict__ vecB,
                            float* __restrict__ out, int n) {
  __shared__ float red[1024];
  const int tid = threadIdx.x;
  float s = 0.f;
  for (int i = tid; i < n; i += 1024) {
    float e = FT[i] + vecB[i];
    s += e * e;
  }
  red[tid] = s; __syncthreads();
  for (int st = 512; st > 0; st >>= 1) {
    if (tid < st) red[tid] += red[tid + st];
    __syncthreads();
  }
  if (tid == 0) out[0] = red[0];
}

// ---------------------------------------------------------------------------
extern "C" void kernel_launch(void* const* d_in, const int* in_sizes, int n_in,
                              void* d_out, int out_size, void* d_ws,
                              size_t ws_size, hipStream_t stream) {
  (void)in_sizes; (void)n_in; (void)out_size; (void)ws_size;
  const float* pts    = (const float*)d_in[1];
  const float* W1     = (const float*)d_in[2];
  const float* b1     = (const float*)d_in[3];
  const float* W2     = (const float*)d_in[4];
  const float* b2     = (const float*)d_in[5];
  const float* W3     = (const float*)d_in[6];
  const float* coeffs = (const float*)d_in[9];
  const float* vecB   = (const float*)d_in[10];

  // workspace layout (buffer lifetimes allow aggressive reuse)
  const size_t BIG = (size_t)NPTS * HDIM * sizeof(float); // 256 MB
  char* w = (char*)d_ws;
  float* h1   = (float*)(w);            // -> reused as du_t after GEMM1
  float* dh1  = (float*)(w + BIG);      // -> reused as mat_A after GEMM1
  float* dh2  = (float*)(w + 2 * BIG);
  float* AtA  = (float*)(w + 3 * BIG);  // 64 MB (fits L2 for CG sweeps)
  char*  tail = w + 3 * BIG + (size_t)NNDIM * NNDIM * sizeof(float);
  float* du_t = h1;
  float* matA = dh1;
  float* rhs  = (float*)tail;
  float* xv   = rhs + NNDIM;
  float* rv   = xv + NNDIM;
  float* pv   = rv + NNDIM;
  float* Apv  = pv + NNDIM;
  float* FT   = Apv + NNDIM;
  float* sc   = FT + NMODES;

  // 1. layer-1 activations + JVP tangents
  {
    long long total = (long long)NPTS * HDIM;
    act_kernel<<<(unsigned)(total / 256), 256, 0, stream>>>(pts, W1, b1, h1, dh1);
  }
  // 2. fused dual GEMM + tanh' epilogue: dh2 = (1-tanh(h1@W2+b2)^2)*(dh1@W2)
  gemm_dual_tanh<<<dim3(HDIM / 64, NPTS / 128), 256, 0, stream>>>(
      h1, dh1, W2, b2, dh2, NPTS, HDIM, HDIM);
  // 3. du_t = dh2 @ W3   (tangent of affine layer; b3 drops out)
  gemm_wmma<false, false, false><<<dim3(NNDIM / 64, NPTS / 128), 256, 0, stream>>>(
      dh2, W3, du_t, nullptr, 0.f, NPTS, NNDIM, HDIM, HDIM, NNDIM);
  // 4. mat_A = diag(coeffs) * DCT @ du_t
  gemm_wmma<false, true, false><<<dim3(NNDIM / 64, NMODES / 128), 256, 0, stream>>>(
      (const float*)d_in[8], du_t, matA, coeffs, 0.f, NMODES, NNDIM, NPTS,
      NPTS, NNDIM);
  // 5. AtA = mat_A^T @ mat_A + regul*I
  gemm_wmma<true, false, true><<<dim3(NNDIM / 64, NNDIM / 128), 256, 0, stream>>>(
      matA, matA, AtA, nullptr, REGUL_L2, NNDIM, NNDIM, NMODES, NNDIM, NNDIM);
  // 6. rhs = mat_A^T @ (-vec_B)
  rhs_kernel<<<NNDIM / 256, 256, 0, stream>>>(matA, vecB, rhs);
  // 7. CG solve (AtA is SPD; fixed iteration count, deterministic)
  cg_init_kernel<<<1, 1024, 0, stream>>>(rhs, xv, rv, pv, sc, NNDIM);
  for (int it = 0; it < CG_ITERS; ++it) {
    gemv_kernel<<<NNDIM / 8, 256, 0, stream>>>(AtA, pv, Apv, NNDIM, NNDIM);
    cg_step_kernel<<<1, 1024, 0, stream>>>(Apv, xv, rv, pv, sc, NNDIM);
  }
  // 8. FT_high = mat_A @ w ; loss = sum((FT + vec_B)^2)
  gemv_kernel<<<NMODES / 8, 256, 0, stream>>>(matA, xv, FT, NMODES, NNDIM);
  loss_kernel<<<1, 1024, 0, stream>>>(FT, vecB, (float*)d_out, NMODES);
}